// SobelLoss_48232482734501
// MI455X (gfx1250) — compile-verified
//
#include <hip/hip_runtime.h>
#include <hip/hip_bf16.h>

// Problem constants from the reference: (B,T,H,W,F) = (2,6,512,512,8)
#define H_IMG 512
#define W_IMG 512
#define FCH   8
#define NPLANE 12                       // B*T planes
#define PLANE_ELEMS (H_IMG * W_IMG * FCH)

#define TILE 16
#define HALO (TILE + 2)                 // 18
#define ROWF (HALO * FCH)               // 144 elements per LDS halo row
#define NCHUNK (HALO * HALO * 2)        // 648 float4 chunks per buffer

// Select data-movement path:
//  1 = clang builtin async global->LDS (confirmed present + codegen-verified on this toolchain)
//  2 = inline-asm global_load_async_to_lds_b128 (fallback for other gfx1250 toolchains)
//  0 = plain vector copy fallback (host pass / other targets)
#if defined(__AMDGCN__) && __has_builtin(__builtin_amdgcn_global_load_async_to_lds_b128)
  #define ASYNC_MODE 1
#elif defined(__gfx1250__)
  #define ASYNC_MODE 2
#else
  #define ASYNC_MODE 0
#endif

#if ASYNC_MODE
// Builtin signature (from clang diagnostic): (v4i AS1* src, v4i AS3* dst, int off, int cpol)
typedef int v4i_ __attribute__((ext_vector_type(4)));
typedef __attribute__((address_space(1))) v4i_* gv4i;
typedef __attribute__((address_space(3))) v4i_* lv4i;

__device__ __forceinline__ void async_wait0() {
#if __has_builtin(__builtin_amdgcn_s_wait_asynccnt)
  __builtin_amdgcn_s_wait_asynccnt(0);
#else
  asm volatile("s_wait_asynccnt 0" ::: "memory");
#endif
}
#endif

__global__ __launch_bounds__(256) void
sobel_l1_kernel(const float* __restrict__ pred, const float* __restrict__ tgt,
                const int* __restrict__ msk, float* __restrict__ out)
{
  __shared__ __align__(16) float sp[HALO * ROWF];   // pred halo tile
  __shared__ __align__(16) float st[HALO * ROWF];   // target halo tile
  __shared__ __align__(16) int   sm[HALO * ROWF];   // mask halo tile

  const int tid = threadIdx.x;
  const int h0 = blockIdx.y * TILE;
  const int w0 = blockIdx.x * TILE;
  const size_t planeOff = (size_t)blockIdx.z * PLANE_ELEMS;
  const float* predP = pred + planeOff;
  const float* tgtP  = tgt  + planeOff;
  const int*   mskP  = msk  + planeOff;

  // ---------------- halo load: 18x18 pixels x 8 ch, float4 chunks ----------------
  for (int c = tid; c < NCHUNK; c += 256) {
    const int p    = c >> 1;
    const int part = (c & 1) * 4;           // which half of the 8-channel pixel
    const int hh   = p / HALO;
    const int ww   = p - hh * HALO;
    const int h    = h0 - 1 + hh;
    const int w    = w0 - 1 + ww;
    const int loff = hh * ROWF + ww * FCH + part;   // element offset in LDS tile
    if ((unsigned)h < (unsigned)H_IMG && (unsigned)w < (unsigned)W_IMG) {
      const int goff = (h * W_IMG + w) * FCH + part;  // element offset in plane
#if ASYNC_MODE == 1
      __builtin_amdgcn_global_load_async_to_lds_b128(
          (gv4i)const_cast<float*>(predP + goff), (lv4i)&sp[loff], 0, 0);
      __builtin_amdgcn_global_load_async_to_lds_b128(
          (gv4i)const_cast<float*>(tgtP + goff),  (lv4i)&st[loff], 0, 0);
      __builtin_amdgcn_global_load_async_to_lds_b128(
          (gv4i)const_cast<int*>(mskP + goff),    (lv4i)&sm[loff], 0, 0);
#elif ASYNC_MODE == 2
      {
        unsigned lp = (unsigned)(size_t)(lv4i)&sp[loff];
        unsigned lt = (unsigned)(size_t)(lv4i)&st[loff];
        unsigned lm = (unsigned)(size_t)(lv4i)&sm[loff];
        unsigned go = (unsigned)(goff * 4);  // byte offset from plane base (< 2^31)
        asm volatile("global_load_async_to_lds_b128 %0, %1, %2 offset:0"
                     :: "v"(lp), "v"(go), "s"(predP) : "memory");
        asm volatile("global_load_async_to_lds_b128 %0, %1, %2 offset:0"
                     :: "v"(lt), "v"(go), "s"(tgtP) : "memory");
        asm volatile("global_load_async_to_lds_b128 %0, %1, %2 offset:0"
                     :: "v"(lm), "v"(go), "s"(mskP) : "memory");
      }
#else
      *(float4*)&sp[loff] = *(const float4*)(predP + goff);
      *(float4*)&st[loff] = *(const float4*)(tgtP + goff);
      *(int4*)&sm[loff]   = *(const int4*)(mskP + goff);
#endif
    } else {
      // SAME zero padding: out-of-image halo is zero (also forces border invalid)
      *(float4*)&sp[loff] = make_float4(0.f, 0.f, 0.f, 0.f);
      *(float4*)&st[loff] = make_float4(0.f, 0.f, 0.f, 0.f);
      *(int4*)&sm[loff]   = make_int4(0, 0, 0, 0);
    }
  }
#if ASYNC_MODE
  async_wait0();          // drain this wave's ASYNCcnt before the barrier
#endif
  __syncthreads();

  // ---------------- stencil compute: 8 outputs per thread ----------------
  const int f  = tid & (FCH - 1);
  const int wl = (tid >> 3) & (TILE - 1);
  const int hs = tid >> 7;                 // 0 or 1: which half of the rows

#pragma unroll
  for (int hl = hs; hl < TILE; hl += 2) {
    const int hc = hl + 1;
    const int wc = wl + 1;
#define AT(dh, dw) ((hc + (dh)) * ROWF + (wc + (dw)) * FCH + f)
    const int msum = sm[AT(-1,-1)] + sm[AT(-1,0)] + sm[AT(-1,1)]
                   + sm[AT( 0,-1)] + sm[AT( 0,0)] + sm[AT( 0,1)]
                   + sm[AT( 1,-1)] + sm[AT( 1,0)] + sm[AT( 1,1)];

    const float pmm = sp[AT(-1,-1)], pmz = sp[AT(-1,0)], pmp = sp[AT(-1,1)];
    const float pzm = sp[AT( 0,-1)],                     pzp = sp[AT( 0,1)];
    const float ppm = sp[AT( 1,-1)], ppz = sp[AT( 1,0)], ppp = sp[AT( 1,1)];
    const float qmm = st[AT(-1,-1)], qmz = st[AT(-1,0)], qmp = st[AT(-1,1)];
    const float qzm = st[AT( 0,-1)],                     qzp = st[AT( 0,1)];
    const float qpm = st[AT( 1,-1)], qpz = st[AT( 1,0)], qpp = st[AT( 1,1)];
#undef AT
    // Sobel (correlation, no flip): gx = right column - left column, gy = bottom - top
    const float gxp = (pmp + 2.f * pzp + ppp) - (pmm + 2.f * pzm + ppm);
    const float gyp = (ppm + 2.f * ppz + ppp) - (pmm + 2.f * pmz + pmp);
    const float gxt = (qmp + 2.f * qzp + qpp) - (qmm + 2.f * qzm + qpm);
    const float gyt = (qpm + 2.f * qpz + qpp) - (qmm + 2.f * qmz + qmp);

    const float loss = (msum == 9) ? (fabsf(gxp - gxt) + fabsf(gyp - gyt)) : 0.f;
    // Output is written once and never re-read: store non-temporal (TH=NT) so the
    // 100 MB result stream doesn't evict reusable input halo lines from L2.
    __builtin_nontemporal_store(
        loss, &out[planeOff + ((size_t)(h0 + hl) * W_IMG + (w0 + wl)) * FCH + f]);
  }
}

extern "C" void kernel_launch(void* const* d_in, const int* in_sizes, int n_in,
                              void* d_out, int out_size, void* d_ws, size_t ws_size,
                              hipStream_t stream) {
  const float* pred = (const float*)d_in[0];
  const float* tgt  = (const float*)d_in[1];
  const int*   msk  = (const int*)d_in[2];
  float* out = (float*)d_out;

  dim3 grid(W_IMG / TILE, H_IMG / TILE, NPLANE);   // 32 x 32 x 12
  dim3 block(256);                                  // 8 waves (wave32)
  hipLaunchKernelGGL(sobel_l1_kernel, grid, block, 0, stream, pred, tgt, msk, out);
}